// DeformConv2d_8383776161842
// MI455X (gfx1250) — compile-verified
//
#include <hip/hip_runtime.h>
#include <math.h>

typedef __attribute__((ext_vector_type(2))) float v2f;
typedef __attribute__((ext_vector_type(8))) float v8f;

#define BSZ   4
#define IC    64
#define OC    128
#define H     128
#define W     128
#define OFFC  18      // 2*N offset channels
#define NPTS  9
#define HP    130     // padded H
#define WP    130
#define XH    384     // expanded x_off spatial size (128*3)
#define OH    382     // final conv output (384 - 3 + 1)
#define OW    382
#define KTOT  (IC*9)                 // 576
#define MTOT  (BSZ*OH*OW)            // 583,696 = 16 * 36,481
#define MTILES (MTOT/16)             // 36,481 (odd)

__device__ __forceinline__ int clampi(int v, int lo, int hi) {
    return v < lo ? lo : (v > hi ? hi : v);
}
__device__ __forceinline__ float clampf(float v, float lo, float hi) {
    return fminf(fmaxf(v, lo), hi);
}
// Read from the zero-padded image xp[b][c][y][x] (pad=1) via the raw x tensor.
__device__ __forceinline__ float sample_pad(const float* __restrict__ x, int b, int c, int y, int xc) {
    int yy = y - 1, xx = xc - 1;
    if ((unsigned)yy < (unsigned)H && (unsigned)xx < (unsigned)W)
        return x[(((size_t)b * IC + c) * H + yy) * W + xx];
    return 0.0f;
}

// ---------------------------------------------------------------------------
// K0: pack w_conv (oc, ic, 3, 3) -> wp[p][oc] = {w[k=2p][oc], w[k=2p+1][oc]}
//     with k = (ky*3+kx)*64 + ic.  One aligned b64 per B-fragment lane-load.
// ---------------------------------------------------------------------------
__global__ void pack_w_kernel(const float* __restrict__ w_conv, float* __restrict__ wp) {
    int i = blockIdx.x * 256 + threadIdx.x;      // over (KTOT/2)*OC = 36,864 pairs
    if (i >= (KTOT / 2) * OC) return;
    int oc = i % OC;
    int p  = i / OC;
    int k0 = 2 * p;
    int ic = k0 % IC;                 // even, ic+1 < IC, same (ky,kx) group
    int kp = k0 / IC;
    int ky = kp / 3, kx = kp % 3;
    const float* wsrc = w_conv + (((size_t)oc * IC + ic) * 3 + ky) * 3 + kx;
    v2f pr;
    pr[0] = wsrc[0];
    pr[1] = wsrc[9];                  // ic+1 => +3*3 floats
    *(v2f*)(wp + (size_t)i * 2) = pr;
}

// ---------------------------------------------------------------------------
// K1: offset conv  (18 out ch, 64 in ch, 3x3, pad 1) + bias
// offs layout: [b][ch][oy][ox]
// ---------------------------------------------------------------------------
__global__ void offset_conv_kernel(const float* __restrict__ x, const float* __restrict__ w_p,
                                   const float* __restrict__ b_p, float* __restrict__ offs) {
    int i = blockIdx.x * 256 + threadIdx.x;     // exact: BSZ*OFFC*H*W threads
    int ox = i % W;  int t = i / W;
    int oy = t % H;  t /= H;
    int ch = t % OFFC;
    int b  = t / OFFC;

    float acc = b_p[ch];
    for (int ky = 0; ky < 3; ++ky) {
        int y = oy + ky - 1;
        if ((unsigned)y >= (unsigned)H) continue;
        for (int kx = 0; kx < 3; ++kx) {
            int xx = ox + kx - 1;
            if ((unsigned)xx >= (unsigned)W) continue;
            const float* xp = x + ((size_t)b * IC * H * W) + (size_t)y * W + xx;
            const float* wpp = w_p + ((size_t)ch * IC * 9) + ky * 3 + kx;
            #pragma unroll 8
            for (int ic = 0; ic < IC; ++ic)
                acc = fmaf(xp[(size_t)ic * H * W], wpp[(size_t)ic * 9], acc);
        }
    }
    offs[i] = acc;
}

// ---------------------------------------------------------------------------
// K2: deformable bilinear sampling -> x_off, NHWC layout:
//     xoff[b][row][col][c], row = oy*3 + (n/3), col = ox*3 + (n%3)
// One wave per (b, oy, ox, n) task; each lane handles 2 channels.
// ---------------------------------------------------------------------------
__global__ void deform_sample_kernel(const float* __restrict__ x, const float* __restrict__ offs,
                                     float* __restrict__ xoff) {
    int wave = threadIdx.x >> 5;
    int lane = threadIdx.x & 31;
    int task = blockIdx.x * 8 + wave;           // exact: BSZ*H*W*9 tasks
    int n  = task % NPTS;  int t = task / NPTS;
    int ox = t % W;        t /= W;
    int oy = t % H;
    int b  = t / H;
    int a  = n / 3;        // row tap index
    int c2 = n % 3;        // col tap index

    float off_r = offs[(((size_t)b * OFFC +     n) * H + oy) * W + ox];
    float off_c = offs[(((size_t)b * OFFC + 9 + n) * H + oy) * W + ox];
    float pr = (float)(oy + a) + off_r;         // (oy+1) + (a-1) + off
    float pc = (float)(ox + c2) + off_c;

    int fr = (int)floorf(pr), fc = (int)floorf(pc);
    int lt_r = clampi(fr,     0, HP - 1), lt_c = clampi(fc,     0, WP - 1);
    int rb_r = clampi(fr + 1, 0, HP - 1), rb_c = clampi(fc + 1, 0, WP - 1);
    float prc = clampf(pr, 0.0f, (float)(HP - 1));
    float pcc = clampf(pc, 0.0f, (float)(WP - 1));

    float g_lt = (1.0f + ((float)lt_r - prc)) * (1.0f + ((float)lt_c - pcc));
    float g_rb = (1.0f - ((float)rb_r - prc)) * (1.0f - ((float)rb_c - pcc));
    float g_lb = (1.0f + ((float)lt_r - prc)) * (1.0f - ((float)rb_c - pcc));
    float g_rt = (1.0f - ((float)rb_r - prc)) * (1.0f + ((float)lt_c - pcc));

    int row = oy * 3 + a, col = ox * 3 + c2;
    float* dst = xoff + (((size_t)b * XH + row) * XH + col) * IC;

    int c0 = lane * 2;
    v2f v;
    #pragma unroll
    for (int j = 0; j < 2; ++j) {
        int c = c0 + j;
        float v_lt = sample_pad(x, b, c, lt_r, lt_c);
        float v_rb = sample_pad(x, b, c, rb_r, rb_c);
        float v_lb = sample_pad(x, b, c, lt_r, rb_c);
        float v_rt = sample_pad(x, b, c, rb_r, lt_c);
        v[j] = g_lt * v_lt + g_rb * v_rb + g_lb * v_lb + g_rt * v_rt;
    }
    *(v2f*)(dst + c0) = v;   // 8B aligned (c0 even, pixel base multiple of 64 floats)
}

// ---------------------------------------------------------------------------
// K3: final 3x3 conv as implicit GEMM with V_WMMA_F32_16X16X4_F32.
// M = BSZ*OH*OW, N = OC, K = IC*9.
// 2x4 register tiling: each wave computes 32M x 64N (8 accumulators);
// per K-step of 4: 2 A b64 + 4 B b64 loads feed 8 wmma -> 0.75 loads/wmma,
// 8 independent accumulator chains for WMMA ILP.
// Block = 8 waves: 2-way split on N-quad, 4-way on M-pair -> 128M x 128N.
// ---------------------------------------------------------------------------
__global__ void __launch_bounds__(256) wmma_conv_kernel(const float* __restrict__ xoff,
                                                        const float* __restrict__ wp,
                                                        float* __restrict__ out) {
    int wave = threadIdx.x >> 5;
    int lane = threadIdx.x & 31;
    int lm   = lane & 15;
    int khalf = (lane >> 4) << 1;           // 0 or 2 (K sub-offset for A/B fragments)

    int n0    = (wave & 1) * 64;            // four N tiles: n0 + {0,16,32,48}
    int nlane = n0 + lm;

    int t0  = blockIdx.x * 8 + (wave >> 1) * 2;   // m-tile indices (clamped: dup tail)
    int mt0 = clampi(t0,     0, MTILES - 1) * 16;
    int mt1 = clampi(t0 + 1, 0, MTILES - 1) * 16;

    // Per-lane A rows for both M tiles
    int mr0 = mt0 + lm;
    int b0  = mr0 / (OH * OW); int r0 = mr0 % (OH * OW);
    int oy0 = r0 / OW,  ox0 = r0 % OW;
    int mr1 = mt1 + lm;
    int b1  = mr1 / (OH * OW); int r1 = mr1 % (OH * OW);
    int oy1 = r1 / OW,  ox1 = r1 % OW;

    v8f acc00 = {}, acc01 = {}, acc02 = {}, acc03 = {};
    v8f acc10 = {}, acc11 = {}, acc12 = {}, acc13 = {};

    #pragma unroll 1
    for (int kp = 0; kp < 9; ++kp) {
        int ky = kp / 3, kx = kp % 3;
        const float* pix0 = xoff + (((size_t)b0 * XH + (oy0 + ky)) * XH + (ox0 + kx)) * IC + khalf;
        const float* pix1 = xoff + (((size_t)b1 * XH + (oy1 + ky)) * XH + (ox1 + kx)) * IC + khalf;
        // wp float2 pair index = (kp*64 + ic4 + khalf)/2 ; float offset = (pair*OC + n)*2
        const float* wb = wp + ((size_t)(kp * 32 + (khalf >> 1)) * OC + nlane) * 2;
        #pragma unroll
        for (int ic4 = 0; ic4 < IC; ic4 += 4) {
            v2f a0  = *(const v2f*)(pix0 + ic4);                  // A tile0 [m][k..k+1]
            v2f a1  = *(const v2f*)(pix1 + ic4);                  // A tile1
            const float* wbk = wb + (size_t)ic4 * OC;
            v2f bv0 = *(const v2f*)(wbk);                         // B ntile n0
            v2f bv1 = *(const v2f*)(wbk + 32);                    // B ntile n0+16
            v2f bv2 = *(const v2f*)(wbk + 64);                    // B ntile n0+32
            v2f bv3 = *(const v2f*)(wbk + 96);                    // B ntile n0+48
            acc00 = __builtin_amdgcn_wmma_f32_16x16x4_f32(false, a0, false, bv0, (short)0, acc00, false, false);
            acc01 = __builtin_amdgcn_wmma_f32_16x16x4_f32(false, a0, false, bv1, (short)0, acc01, false, false);
            acc02 = __builtin_amdgcn_wmma_f32_16x16x4_f32(false, a0, false, bv2, (short)0, acc02, false, false);
            acc03 = __builtin_amdgcn_wmma_f32_16x16x4_f32(false, a0, false, bv3, (short)0, acc03, false, false);
            acc10 = __builtin_amdgcn_wmma_f32_16x16x4_f32(false, a1, false, bv0, (short)0, acc10, false, false);
            acc11 = __builtin_amdgcn_wmma_f32_16x16x4_f32(false, a1, false, bv1, (short)0, acc11, false, false);
            acc12 = __builtin_amdgcn_wmma_f32_16x16x4_f32(false, a1, false, bv2, (short)0, acc12, false, false);
            acc13 = __builtin_amdgcn_wmma_f32_16x16x4_f32(false, a1, false, bv3, (short)0, acc13, false, false);
        }
    }

    // D tiles: element (M = r + 8*(lane>=16), N = lane&15) lives in acc[r]
    int half8 = (lane >> 4) << 3;
    auto store_tile = [&](int mt, int nl, const v8f& acc) {
        #pragma unroll
        for (int r = 0; r < 8; ++r) {
            int m   = mt + r + half8;
            int b2  = m / (OH * OW);
            int rm  = m % (OH * OW);
            int oy2 = rm / OW;
            int ox2 = rm % OW;
            out[((((size_t)b2 * OC) + nl) * OH + oy2) * OW + ox2] = acc[r];
        }
    };
    store_tile(mt0, nlane,      acc00);
    store_tile(mt0, nlane + 16, acc01);
    store_tile(mt0, nlane + 32, acc02);
    store_tile(mt0, nlane + 48, acc03);
    store_tile(mt1, nlane,      acc10);
    store_tile(mt1, nlane + 16, acc11);
    store_tile(mt1, nlane + 32, acc12);
    store_tile(mt1, nlane + 48, acc13);
}

// ---------------------------------------------------------------------------
extern "C" void kernel_launch(void* const* d_in, const int* in_sizes, int n_in,
                              void* d_out, int out_size, void* d_ws, size_t ws_size,
                              hipStream_t stream) {
    const float* x      = (const float*)d_in[0];
    const float* w_p    = (const float*)d_in[1];
    const float* b_p    = (const float*)d_in[2];
    const float* w_conv = (const float*)d_in[3];
    float* out = (float*)d_out;

    float* offs = (float*)d_ws;                               // 4*18*128*128   = 1,179,648 f
    float* wp   = offs + (size_t)BSZ * OFFC * H * W;          // 576*128        = 73,728 f
    float* xoff = wp + (size_t)KTOT * OC;                     // 4*384*384*64   = 37,748,736 f

    pack_w_kernel<<<((KTOT / 2) * OC + 255) / 256, 256, 0, stream>>>(w_conv, wp);
    offset_conv_kernel<<<(BSZ * OFFC * H * W) / 256, 256, 0, stream>>>(x, w_p, b_p, offs);
    deform_sample_kernel<<<(BSZ * H * W * NPTS) / 8, 256, 0, stream>>>(x, offs, xoff);
    wmma_conv_kernel<<<(MTILES + 7) / 8, 256, 0, stream>>>(xoff, wp, out);
}